// MultiTeacherOperation_11364483465328
// MI455X (gfx1250) — compile-verified
//
#include <hip/hip_runtime.h>
#include <math.h>

#define EPS_ 1e-5f
#define B_   16
#define N_   2048
#define S_   32
#define KNN_ 12

typedef __attribute__((ext_vector_type(2))) float v2f;
typedef __attribute__((ext_vector_type(4))) float v4f;
typedef __attribute__((ext_vector_type(8))) float v8f;

// ---------------------------------------------------------------------------
// 1) Furthest-point sampling on xyz_t1 (reference semantics: start at idx 0,
//    argmax breaks ties toward the lowest index). One block per batch.
// ---------------------------------------------------------------------------
__global__ void fps_kernel(const float* __restrict__ xyz,
                           int* __restrict__ fps_idx,
                           float* __restrict__ new_xyz) {
  const int b = blockIdx.x;
  const int t = threadIdx.x;            // 256 threads
  __shared__ float d[N_];
  __shared__ float red_d[256];
  __shared__ int   red_i[256];
  __shared__ float sel[3];
  const float* xb = xyz + (size_t)b * N_ * 3;

  for (int j = t; j < N_; j += 256) d[j] = 1e10f;
  if (t == 0) {
    fps_idx[b * S_ + 0] = 0;
    sel[0] = xb[0]; sel[1] = xb[1]; sel[2] = xb[2];
  }
  __syncthreads();

  for (int it = 1; it < S_; ++it) {
    const float px = sel[0], py = sel[1], pz = sel[2];
    float bd = -1.0f; int bi = 0x7fffffff;
    for (int j = t; j < N_; j += 256) {
      float dx = xb[j * 3 + 0] - px;
      float dy = xb[j * 3 + 1] - py;
      float dz = xb[j * 3 + 2] - pz;
      float nd = fminf(d[j], dx * dx + dy * dy + dz * dz);
      d[j] = nd;
      if (nd > bd) { bd = nd; bi = j; }   // ascending j -> first max kept
    }
    red_d[t] = bd; red_i[t] = bi;
    __syncthreads();
    for (int off = 128; off > 0; off >>= 1) {
      if (t < off) {
        float od = red_d[t + off]; int oi = red_i[t + off];
        if (od > red_d[t] || (od == red_d[t] && oi < red_i[t])) {
          red_d[t] = od; red_i[t] = oi;
        }
      }
      __syncthreads();
    }
    if (t == 0) {
      int w = red_i[0];
      fps_idx[b * S_ + it] = w;
      sel[0] = xb[w * 3 + 0]; sel[1] = xb[w * 3 + 1]; sel[2] = xb[w * 3 + 2];
    }
    __syncthreads();
  }
  if (t < S_) {
    int ii = fps_idx[b * S_ + t];
    new_xyz[(b * S_ + t) * 3 + 0] = xb[ii * 3 + 0];
    new_xyz[(b * S_ + t) * 3 + 1] = xb[ii * 3 + 1];
    new_xyz[(b * S_ + t) * 3 + 2] = xb[ii * 3 + 2];
  }
}

// ---------------------------------------------------------------------------
// 2) kNN (k=12, squared distance) of the 32 sampled centers into each of the
//    three clouds. One 64-thread block per (query, cloud).
// ---------------------------------------------------------------------------
__global__ void knn_kernel(const float* __restrict__ xyz_t1,
                           const float* __restrict__ xyz_t2,
                           const float* __restrict__ xyz_s,
                           const float* __restrict__ new_xyz,
                           int* __restrict__ idx_t1,
                           int* __restrict__ idx_t2,
                           int* __restrict__ idx_s) {
  const int q = blockIdx.x;       // 0..B*S-1
  const int which = blockIdx.y;   // 0: t1, 1: t2, 2: s
  const int t = threadIdx.x;      // 64 threads
  const float* xyz = (which == 0) ? xyz_t1 : (which == 1) ? xyz_t2 : xyz_s;
  int* out = (which == 0) ? idx_t1 : (which == 1) ? idx_t2 : idx_s;
  const int b = q >> 5;
  const float* xb = xyz + (size_t)b * N_ * 3;
  const float qx = new_xyz[q * 3 + 0];
  const float qy = new_xyz[q * 3 + 1];
  const float qz = new_xyz[q * 3 + 2];

  float dl[KNN_]; int il[KNN_];
#pragma unroll
  for (int i = 0; i < KNN_; ++i) { dl[i] = 3.0e38f; il[i] = 0; }

  for (int p = t; p < N_; p += 64) {     // 32 candidates per thread (>= 12)
    float dx = xb[p * 3 + 0] - qx;
    float dy = xb[p * 3 + 1] - qy;
    float dz = xb[p * 3 + 2] - qz;
    float dd = dx * dx + dy * dy + dz * dz;
    if (dd < dl[KNN_ - 1]) {
      int pos = KNN_ - 1;
      while (pos > 0 && dl[pos - 1] > dd) {
        dl[pos] = dl[pos - 1]; il[pos] = il[pos - 1]; --pos;
      }
      dl[pos] = dd; il[pos] = p;
    }
  }

  __shared__ float md[64 * KNN_];
  __shared__ int   mi[64 * KNN_];
  __shared__ int   head[64];
#pragma unroll
  for (int i = 0; i < KNN_; ++i) { md[t * KNN_ + i] = dl[i]; mi[t * KNN_ + i] = il[i]; }
  head[t] = 0;
  __syncthreads();

  if (t == 0) {                          // 12-way merge of 64 sorted lists
    for (int j = 0; j < KNN_; ++j) {
      float best = 3.4e38f; int bt = 0;
      for (int w = 0; w < 64; ++w) {
        float c = (head[w] < KNN_) ? md[w * KNN_ + head[w]] : 3.4e38f;
        if (c < best) { best = c; bt = w; }
      }
      out[q * KNN_ + j] = mi[bt * KNN_ + head[bt]];
      head[bt]++;
    }
  }
}

// ---------------------------------------------------------------------------
// 3) Fused gather -> GEMM (fp32 WMMA 16x16x4) -> +bias -> BN -> ReLU -> max(12)
//    A: [6144, K] gathered rows, B: W^T (W is [1024, K] row-major).
//    Block: 128 threads (4 waves); tile 48 rows (4 groups of 12) x 64 cols.
// ---------------------------------------------------------------------------
#define MT  48
#define NT  64
#define KT  32
#define LDT 36   // padded stride: conflict-free frag reads, 16B-aligned stores

__global__ __launch_bounds__(128) void graph_gemm_kernel(
    const float* __restrict__ feat, const int* __restrict__ idx,
    const float* __restrict__ W, const float* __restrict__ bias,
    const float* __restrict__ bn_g, const float* __restrict__ bn_b,
    const float* __restrict__ bn_m, const float* __restrict__ bn_v,
    float* __restrict__ out, int K) {
  __shared__ float As[MT * LDT];
  __shared__ float Ws[NT * LDT];
  __shared__ float Cs[MT * NT];
  __shared__ unsigned rowOff[MT];

  const int t    = threadIdx.x;
  const int wave = t >> 5;
  const int lane = t & 31;
  const int hf   = lane >> 4;   // half-wave select (K sub-column per ISA layout)
  const int l16  = lane & 15;
  const int row0 = blockIdx.x * MT;
  const int col0 = blockIdx.y * NT;

  if (t < MT) {
    int m = row0 + t;
    int bb = m / (S_ * KNN_);            // 384 rows per batch
    rowOff[t] = (unsigned)((bb * N_ + idx[m]) * K);
  }
  __syncthreads();

  v8f acc0 = {}, acc1 = {}, acc2 = {};

  for (int k0 = 0; k0 < K; k0 += KT) {
    // A tile 48x32: 384 float4, 3 per thread (gathered rows)
#pragma unroll
    for (int i = 0; i < 3; ++i) {
      int qq = t + i * 128;
      int r = qq >> 3, c4 = (qq & 7) << 2;
      v4f v = *(const v4f*)(feat + rowOff[r] + k0 + c4);
      *(v4f*)(&As[r * LDT + c4]) = v;
    }
    // B tile (W^T) 64 cols x 32 k: 512 float4, 4 per thread
#pragma unroll
    for (int i = 0; i < 4; ++i) {
      int qq = t + i * 128;
      int n = qq >> 3, c4 = (qq & 7) << 2;
      v4f v = *(const v4f*)(W + (size_t)(col0 + n) * K + k0 + c4);
      *(v4f*)(&Ws[n * LDT + c4]) = v;
    }
    __syncthreads();

#pragma unroll
    for (int kk = 0; kk < KT; kk += 4) {
      int ke = kk + 2 * hf;
      v2f bf = *(const v2f*)(&Ws[(wave * 16 + l16) * LDT + ke]);
      v2f a0 = *(const v2f*)(&As[(0  + l16) * LDT + ke]);
      v2f a1 = *(const v2f*)(&As[(16 + l16) * LDT + ke]);
      v2f a2 = *(const v2f*)(&As[(32 + l16) * LDT + ke]);
      acc0 = __builtin_amdgcn_wmma_f32_16x16x4_f32(false, a0, false, bf, (short)0, acc0, false, false);
      acc1 = __builtin_amdgcn_wmma_f32_16x16x4_f32(false, a1, false, bf, (short)0, acc1, false, false);
      acc2 = __builtin_amdgcn_wmma_f32_16x16x4_f32(false, a2, false, bf, (short)0, acc2, false, false);
    }
    __syncthreads();
  }

  // Spill C fragments to LDS (ISA 16x16 f32 C layout: m = r + 8*half, n = l16)
#pragma unroll
  for (int r = 0; r < 8; ++r) {
    int mr = r + 8 * hf;
    Cs[(0  + mr) * NT + wave * 16 + l16] = acc0[r];
    Cs[(16 + mr) * NT + wave * 16 + l16] = acc1[r];
    Cs[(32 + mr) * NT + wave * 16 + l16] = acc2[r];
  }
  __syncthreads();

  // Epilogue: y*scale+shift (bias folded into BN), ReLU, max over k=12.
#pragma unroll
  for (int i = 0; i < 2; ++i) {
    int qq = t + i * 128;                 // 4 groups x 64 cols = 256 items
    int g = qq >> 6, c = qq & 63;
    int ch = col0 + c;
    float sc = bn_g[ch] * rsqrtf(bn_v[ch] + EPS_);
    float of = (bias[ch] - bn_m[ch]) * sc + bn_b[ch];
    float best = 0.0f;                    // ReLU outputs are >= 0
#pragma unroll
    for (int kk = 0; kk < KNN_; ++kk) {
      float y = Cs[(g * KNN_ + kk) * NT + c];
      best = fmaxf(best, y * sc + of);
    }
    int G = blockIdx.x * 4 + g;           // (b*32+s)
    out[(size_t)G * 1024 + ch] = best;
  }
}

// ---------------------------------------------------------------------------
// 4) TeacherFusion SE gate: mean over centers, fc1+ReLU, fc2, softmax(2).
// ---------------------------------------------------------------------------
__global__ void fusion_gate_kernel(const float* __restrict__ out_t1,
                                   const float* __restrict__ out_t2,
                                   const float* __restrict__ fc1,
                                   const float* __restrict__ fc2,
                                   float* __restrict__ wgt) {
  const int b = blockIdx.x;
  const int t = threadIdx.x;              // 256
  __shared__ float y[2048];
  __shared__ float h[128];
  const float* p1 = out_t1 + (size_t)b * S_ * 1024;
  const float* p2 = out_t2 + (size_t)b * S_ * 1024;

  for (int c = t; c < 1024; c += 256) {
    float s1 = 0.0f, s2 = 0.0f;
    for (int s = 0; s < S_; ++s) { s1 += p1[s * 1024 + c]; s2 += p2[s * 1024 + c]; }
    y[c]        = s1 * (1.0f / 32.0f);
    y[1024 + c] = s2 * (1.0f / 32.0f);
  }
  __syncthreads();
  if (t < 128) {
    float acc = 0.0f;
    const float* wr = fc1 + (size_t)t * 2048;
    for (int c = 0; c < 2048; ++c) acc += y[c] * wr[c];
    h[t] = fmaxf(acc, 0.0f);
  }
  __syncthreads();
  if (t == 0) {
    float z0 = 0.0f, z1 = 0.0f;
    for (int c = 0; c < 128; ++c) { z0 += h[c] * fc2[c]; z1 += h[c] * fc2[128 + c]; }
    float m = fmaxf(z0, z1);
    float e0 = expf(z0 - m), e1 = expf(z1 - m);
    float inv = 1.0f / (e0 + e1);
    wgt[b * 2 + 0] = e0 * inv;
    wgt[b * 2 + 1] = e1 * inv;
  }
}

// ---------------------------------------------------------------------------
// 5) fused = w0*out_t1 + w1*out_t2  (written straight into d_out segment)
// ---------------------------------------------------------------------------
__global__ void fused_store_kernel(const float* __restrict__ out_t1,
                                   const float* __restrict__ out_t2,
                                   const float* __restrict__ wgt,
                                   float* __restrict__ fused) {
  size_t g = (size_t)blockIdx.x * blockDim.x + threadIdx.x;  // 524288 total
  int b = (int)(g >> 15);                                    // 32*1024 = 2^15
  fused[g] = wgt[b * 2 + 0] * out_t1[g] + wgt[b * 2 + 1] * out_t2[g];
}

// ---------------------------------------------------------------------------
// 6) Classifier: max over centers, two BN+ReLU FC layers, final 15-way FC.
// ---------------------------------------------------------------------------
__global__ void classifier_kernel(
    const float* __restrict__ fused,
    const float* __restrict__ W1, const float* __restrict__ b1,
    const float* __restrict__ g1, const float* __restrict__ bb1,
    const float* __restrict__ m1, const float* __restrict__ v1,
    const float* __restrict__ W2, const float* __restrict__ b2,
    const float* __restrict__ g2, const float* __restrict__ bb2,
    const float* __restrict__ m2, const float* __restrict__ v2,
    const float* __restrict__ W3, const float* __restrict__ b3,
    float* __restrict__ logits) {
  const int b = blockIdx.x, t = threadIdx.x;   // 256
  __shared__ float x[1024];
  __shared__ float x1s[512];
  __shared__ float x2s[256];
  const float* fb = fused + (size_t)b * S_ * 1024;

  for (int c = t; c < 1024; c += 256) {
    float mx = fb[c];
    for (int s = 1; s < S_; ++s) mx = fmaxf(mx, fb[s * 1024 + c]);
    x[c] = mx;
  }
  __syncthreads();
  for (int r = t; r < 512; r += 256) {
    float acc = 0.0f;
    const float* wr = W1 + (size_t)r * 1024;
    for (int c = 0; c < 1024; ++c) acc += x[c] * wr[c];
    float sc = g1[r] * rsqrtf(v1[r] + EPS_);
    x1s[r] = fmaxf((acc + b1[r] - m1[r]) * sc + bb1[r], 0.0f);
  }
  __syncthreads();
  for (int r = t; r < 256; r += 256) {
    float acc = 0.0f;
    const float* wr = W2 + (size_t)r * 512;
    for (int c = 0; c < 512; ++c) acc += x1s[c] * wr[c];
    float sc = g2[r] * rsqrtf(v2[r] + EPS_);
    x2s[r] = fmaxf((acc + b2[r] - m2[r]) * sc + bb2[r], 0.0f);
  }
  __syncthreads();
  if (t < 15) {
    float acc = 0.0f;
    const float* wr = W3 + (size_t)t * 256;
    for (int c = 0; c < 256; ++c) acc += x2s[c] * wr[c];
    logits[b * 15 + t] = acc + b3[t];
  }
}

// ---------------------------------------------------------------------------
// kernel_launch
// Inputs (setup_inputs dict order; params expanded in insertion order):
//  0 feature_s  1 xyz_s  2 feature_t1  3 xyz_t1  4 feature_t2  5 xyz_t2
//  6..11  gc_s_W,  gc_s_bias,  gc_s_bn_{g,b,m,v}
// 12..17  gc_t1_W, gc_t1_bias, gc_t1_bn_{g,b,m,v}
// 18..23  gc_t2_W, gc_t2_bias, gc_t2_bn_{g,b,m,v}
// 24 tf_fc1  25 tf_fc2
// 26..31  cl_W1, cl_b1, cl_bn1_{g,b,m,v}
// 32..37  cl_W2, cl_b2, cl_bn2_{g,b,m,v}
// 38 cl_W3  39 cl_b3
// d_out: out_s [16,32,1024] | fused [16,32,1024] | logits [16,15]
// ---------------------------------------------------------------------------
extern "C" void kernel_launch(void* const* d_in, const int* in_sizes, int n_in,
                              void* d_out, int out_size, void* d_ws, size_t ws_size,
                              hipStream_t stream) {
  (void)in_sizes; (void)n_in; (void)out_size; (void)ws_size;

  const float* feature_s  = (const float*)d_in[0];
  const float* xyz_s      = (const float*)d_in[1];
  const float* feature_t1 = (const float*)d_in[2];
  const float* xyz_t1     = (const float*)d_in[3];
  const float* feature_t2 = (const float*)d_in[4];
  const float* xyz_t2     = (const float*)d_in[5];

  const float* gcs_W  = (const float*)d_in[6];
  const float* gcs_bi = (const float*)d_in[7];
  const float* gcs_g  = (const float*)d_in[8];
  const float* gcs_b  = (const float*)d_in[9];
  const float* gcs_m  = (const float*)d_in[10];
  const float* gcs_v  = (const float*)d_in[11];

  const float* gt1_W  = (const float*)d_in[12];
  const float* gt1_bi = (const float*)d_in[13];
  const float* gt1_g  = (const float*)d_in[14];
  const float* gt1_b  = (const float*)d_in[15];
  const float* gt1_m  = (const float*)d_in[16];
  const float* gt1_v  = (const float*)d_in[17];

  const float* gt2_W  = (const float*)d_in[18];
  const float* gt2_bi = (const float*)d_in[19];
  const float* gt2_g  = (const float*)d_in[20];
  const float* gt2_b  = (const float*)d_in[21];
  const float* gt2_m  = (const float*)d_in[22];
  const float* gt2_v  = (const float*)d_in[23];

  const float* tf_fc1 = (const float*)d_in[24];
  const float* tf_fc2 = (const float*)d_in[25];

  const float* cl_W1 = (const float*)d_in[26];
  const float* cl_b1 = (const float*)d_in[27];
  const float* bn1_g = (const float*)d_in[28];
  const float* bn1_b = (const float*)d_in[29];
  const float* bn1_m = (const float*)d_in[30];
  const float* bn1_v = (const float*)d_in[31];

  const float* cl_W2 = (const float*)d_in[32];
  const float* cl_b2 = (const float*)d_in[33];
  const float* bn2_g = (const float*)d_in[34];
  const float* bn2_b = (const float*)d_in[35];
  const float* bn2_m = (const float*)d_in[36];
  const float* bn2_v = (const float*)d_in[37];

  const float* cl_W3 = (const float*)d_in[38];
  const float* cl_b3 = (const float*)d_in[39];

  // workspace layout (float/int units of 4B)
  float* wsf = (float*)d_ws;
  int*   fps_idx = (int*)d_ws;                 // 512
  float* new_xyz = wsf + 512;                  // 1536
  int*   idx_t1  = (int*)(wsf + 2048);         // 6144
  int*   idx_t2  = (int*)(wsf + 8192);         // 6144
  int*   idx_s   = (int*)(wsf + 14336);        // 6144
  float* wgt     = wsf + 20480;                // 32
  float* out_t1  = wsf + 20992;                // 524288
  float* out_t2  = wsf + 545280;               // 524288  (~4.3 MB total)

  float* out_s_seg  = (float*)d_out;           // 524288
  float* fused_seg  = (float*)d_out + 524288;  // 524288
  float* logits_seg = (float*)d_out + 1048576; // 240

  fps_kernel<<<B_, 256, 0, stream>>>(xyz_t1, fps_idx, new_xyz);

  knn_kernel<<<dim3(B_ * S_, 3), 64, 0, stream>>>(
      xyz_t1, xyz_t2, xyz_s, new_xyz, idx_t1, idx_t2, idx_s);

  dim3 gg(128, 16);   // M=6144/48, N=1024/64
  graph_gemm_kernel<<<gg, 128, 0, stream>>>(
      feature_s, idx_s, gcs_W, gcs_bi, gcs_g, gcs_b, gcs_m, gcs_v,
      out_s_seg, 256);
  graph_gemm_kernel<<<gg, 128, 0, stream>>>(
      feature_t1, idx_t1, gt1_W, gt1_bi, gt1_g, gt1_b, gt1_m, gt1_v,
      out_t1, 1024);
  graph_gemm_kernel<<<gg, 128, 0, stream>>>(
      feature_t2, idx_t2, gt2_W, gt2_bi, gt2_g, gt2_b, gt2_m, gt2_v,
      out_t2, 1024);

  fusion_gate_kernel<<<B_, 256, 0, stream>>>(out_t1, out_t2, tf_fc1, tf_fc2, wgt);

  fused_store_kernel<<<2048, 256, 0, stream>>>(out_t1, out_t2, wgt, fused_seg);

  classifier_kernel<<<B_, 256, 0, stream>>>(
      fused_seg,
      cl_W1, cl_b1, bn1_g, bn1_b, bn1_m, bn1_v,
      cl_W2, cl_b2, bn2_g, bn2_b, bn2_m, bn2_v,
      cl_W3, cl_b3, logits_seg);
}